// Pair_Construct_6451040878733
// MI455X (gfx1250) — compile-verified
//
#include <hip/hip_runtime.h>
#include <hip/hip_bf16.h>
#include <math.h>

// ---------------------------------------------------------------------------
// Problem constants (fixed by reference: B=32, L=128, D=300, P=50, K=12)
// ---------------------------------------------------------------------------
#define BB      32
#define LL      128
#define DD      300
#define PP      50
#define KK      12
#define NPAIR   3044          // sum of per-emotion window sizes, |rel|<=12
#define FF      (2*DD + PP)   // 650
#define NREL    (2*KK + 1)    // 25
#define MROWS   (BB*LL)       // 4096
#define LDS_STR 304           // LDS row stride (floats): even (b64 aligned)

typedef float v2f __attribute__((ext_vector_type(2)));
typedef float v8f __attribute__((ext_vector_type(8)));

// ---------------------------------------------------------------------------
// Kernel 0: build pair table (emo0, cau0, rel+K) in ws and emo_cau_pos floats
// in d_out tail. Order matches reference: emo slowest, cau fastest, pruned.
// ---------------------------------------------------------------------------
__global__ void build_pairs_kernel(int* __restrict__ ptab,
                                   float* __restrict__ out_pos) {
  __shared__ int cnts[LL];
  __shared__ int offs[LL];
  int t = threadIdx.x;            // 0..127 -> emo e = t+1 (1-based)
  int e  = t + 1;
  int lo = (e - KK) < 1 ? 1 : (e - KK);
  int hi = (e + KK) > LL ? LL : (e + KK);
  cnts[t] = hi - lo + 1;
  __syncthreads();
  if (t == 0) {
    int a = 0;
    for (int j = 0; j < LL; ++j) { offs[j] = a; a += cnts[j]; }
  }
  __syncthreads();
  int o = offs[t];
  for (int c = lo; c <= hi; ++c, ++o) {
    ptab[o*3 + 0] = e - 1;            // 0-based emotion index
    ptab[o*3 + 1] = c - 1;            // 0-based cause index
    ptab[o*3 + 2] = (c - e) + KK;     // rel + K in [0,24]
    out_pos[o*2 + 0] = (float)e;      // emo_cau_pos (1-based), exact in fp32
    out_pos[o*2 + 1] = (float)c;
  }
}

// ---------------------------------------------------------------------------
// Kernel 1: pairbase[u][d] = sum_p posf[u][p] * W_hid[d][600+p] + b_hid[d]
// where posf[u][p] = sum_v (128-|v-12|) * exp(-(u-v)^2) * pos_emb[v][p].
// Exploits that pos_feat has only 25 distinct rows (one per rel value).
// ---------------------------------------------------------------------------
__global__ void build_pairbase_kernel(const float* __restrict__ pos_emb,
                                      const float* __restrict__ W_hid,
                                      const float* __restrict__ b_hid,
                                      float* __restrict__ pairbase) {
  __shared__ float posf[NREL * PP];    // 25*50 = 1250 floats
  int tid = threadIdx.x;               // blockDim.x = 256
  for (int idx = tid; idx < NREL * PP; idx += 256) {
    int u = idx / PP, p = idx % PP;
    float acc = 0.f;
    #pragma unroll
    for (int v = 0; v < NREL; ++v) {
      float cnt = (float)(LL - abs(v - KK));
      float d   = (float)(u - v);
      acc += cnt * __expf(-d * d) * pos_emb[v * PP + p];
    }
    posf[idx] = acc;
  }
  __syncthreads();
  for (int idx = tid; idx < NREL * DD; idx += 256) {
    int u = idx / DD, d = idx % DD;
    float acc = b_hid[d];
    const float* wrow = W_hid + d * FF + 2 * DD;   // W3 slice [d][600..649]
    #pragma unroll 10
    for (int p = 0; p < PP; ++p) acc += posf[u * PP + p] * wrow[p];
    pairbase[idx] = acc;
  }
}

// ---------------------------------------------------------------------------
// Kernel 2: dense GEMMs via V_WMMA_F32_16X16X4_F32 with LDS staging.
//   blockIdx.z == 0 : A_out[m][d] = sum_k (h_e+h_share)[m][k] * W_hid[d][k]
//   blockIdx.z == 1 : C_out[m][d] = sum_k (h_c+h_share)[m][k] * W_hid[d][300+k]
// Block = 128 threads (4 waves). Block stages a 64x300 activation tile in LDS
// ONCE, then loops over the 19 N-tiles; each tile stages a zero-padded 16x300
// W slice in LDS. Steady-state k-loop: 2x ds_load_b64 + 1x v_wmma, no EXEC
// divergence (out-of-range columns compute garbage, masked at store).
// Fragment layouts per ISA 7.12.2:
//   A 16x4 : lanes 0-15 hold M=lane, VGPR{0,1}=K{0,1}; lanes 16-31: K{2,3}
//   B 4x16 : VGPR0 = rows K0(l0-15)/K2(l16-31), VGPR1 = K1/K3, N = lane&15
//   C 16x16: acc[j] = element (M = j + 8*(lane>>4), N = lane&15)
// LDS: (64+16)*304*4 = 97,280 B of the WGP's 320 KB.
// ---------------------------------------------------------------------------
__global__ void gemm_wmma_kernel(const float* __restrict__ h_e,
                                 const float* __restrict__ h_c,
                                 const float* __restrict__ h_share,
                                 const float* __restrict__ W_hid,
                                 float* __restrict__ A_out,
                                 float* __restrict__ C_out) {
  __shared__ float sA[64 * LDS_STR];   // 64 activation rows
  __shared__ float sW[16 * LDS_STR];   // 16 weight rows (current N tile)

  const int lane = threadIdx.x & 31;
  const int wave = threadIdx.x >> 5;                 // 4 waves / block
  const int mblk = blockIdx.x * 64;                  // grid.x = 64 -> M=4096
  const int mat  = blockIdx.z;                       // 0: he*W1, 1: hc*W2

  const float* __restrict__ X = mat ? h_c : h_e;
  const int woff = mat ? DD : 0;
  float* __restrict__ O = mat ? C_out : A_out;

  // Stage activations: sA[r][c] = X[mblk+r][c] + h_share[mblk+r][c]
  for (int idx = threadIdx.x; idx < 64 * (DD / 2); idx += 128) {
    const int r  = idx / (DD / 2);
    const int c2 = idx - r * (DD / 2);
    const float2 xv = *((const float2*)(X       + (size_t)(mblk + r) * DD) + c2);
    const float2 sv = *((const float2*)(h_share + (size_t)(mblk + r) * DD) + c2);
    sA[r * LDS_STR + c2 * 2]     = xv.x + sv.x;
    sA[r * LDS_STR + c2 * 2 + 1] = xv.y + sv.y;
  }

  const int mloc = wave * 16 + (lane & 15);          // sA row for A fragment
  const int nloc = lane & 15;                        // sW row for B fragment
  const int kb   = (lane >> 4) * 2;                  // 0 or 2

  for (int n0 = 0; n0 < 19 * 16; n0 += 16) {
    __syncthreads();   // sA ready (1st iter) / previous tile's compute done
    // Stage weight slice rows n0..n0+15 (zero-pad rows >= 300)
    for (int idx = threadIdx.x; idx < 16 * (DD / 2); idx += 128) {
      const int r  = idx / (DD / 2);
      const int c2 = idx - r * (DD / 2);
      const int n  = n0 + r;
      float2 wv = make_float2(0.f, 0.f);
      if (n < DD)
        wv = *((const float2*)(W_hid + (size_t)n * FF + woff) + c2);
      sW[r * LDS_STR + c2 * 2]     = wv.x;
      sW[r * LDS_STR + c2 * 2 + 1] = wv.y;
    }
    __syncthreads();

    v8f acc = {};
    #pragma unroll 5
    for (int k0 = 0; k0 < DD; k0 += 4) {
      const int k = k0 + kb;
      const v2f a = *(const v2f*)&sA[mloc * LDS_STR + k];
      const v2f b = *(const v2f*)&sW[nloc * LDS_STR + k];
      acc = __builtin_amdgcn_wmma_f32_16x16x4_f32(
          /*neg_a=*/false, a, /*neg_b=*/false, b,
          /*c_mod=*/(short)0, acc, /*reuse_a=*/false, /*reuse_b=*/false);
    }

    const int outcol = n0 + nloc;
    if (outcol < DD) {
      const int rbase = mblk + wave * 16 + 8 * (lane >> 4);
      #pragma unroll
      for (int j = 0; j < 8; ++j)
        O[(size_t)(rbase + j) * DD + outcol] = acc[j];
    }
  }
}

// ---------------------------------------------------------------------------
// Kernel 3: per-pair gather + LayerNorm + ELU + dot(W_rel). One wave32 per
// (b, pair). 300 = 9*32 + 12 -> 10 strided elements/lane; butterfly reduce.
// ---------------------------------------------------------------------------
__global__ void pair_score_kernel(const float* __restrict__ A,
                                  const float* __restrict__ C,
                                  const float* __restrict__ pairbase,
                                  const int*  __restrict__ ptab,
                                  const float* __restrict__ ln_g,
                                  const float* __restrict__ ln_b,
                                  const float* __restrict__ W_rel,
                                  const float* __restrict__ b_rel,
                                  float* __restrict__ out) {
  const int gwave = (blockIdx.x * blockDim.x + threadIdx.x) >> 5;
  const int lane  = threadIdx.x & 31;
  if (gwave >= BB * NPAIR) return;
  const int b = gwave / NPAIR;
  const int i = gwave - b * NPAIR;

  const int emo  = ptab[i*3 + 0];
  const int cau  = ptab[i*3 + 1];
  const int relk = ptab[i*3 + 2];

  const float* __restrict__ ar = A + (size_t)(b * LL + emo) * DD;
  const float* __restrict__ cr = C + (size_t)(b * LL + cau) * DD;
  const float* __restrict__ pb = pairbase + relk * DD;

  float hv[10];
  float s = 0.f, s2 = 0.f;
  #pragma unroll
  for (int t = 0; t < 10; ++t) {
    const int d = lane + t * 32;
    float v = 0.f;
    if (d < DD) {
      v = ar[d] + cr[d] + pb[d];
      s += v; s2 += v * v;
    }
    hv[t] = v;
  }
  #pragma unroll
  for (int off = 16; off > 0; off >>= 1) {
    s  += __shfl_xor(s,  off, 32);
    s2 += __shfl_xor(s2, off, 32);
  }
  const float inv_d = 1.f / (float)DD;
  const float mu   = s * inv_d;
  const float var  = s2 * inv_d - mu * mu;
  const float rstd = rsqrtf(var + 1e-5f);

  float dot = 0.f;
  #pragma unroll
  for (int t = 0; t < 10; ++t) {
    const int d = lane + t * 32;
    if (d < DD) {
      float ln  = (hv[t] - mu) * rstd * ln_g[d] + ln_b[d];
      float act = (ln > 0.f) ? ln : (__expf(ln) - 1.f);   // ELU, alpha=1
      dot += act * W_rel[d];
    }
  }
  #pragma unroll
  for (int off = 16; off > 0; off >>= 1) dot += __shfl_xor(dot, off, 32);

  if (lane == 0) out[(size_t)b * NPAIR + i] = dot + b_rel[0];
}

// ---------------------------------------------------------------------------
// Launcher. Inputs (setup_inputs order):
//  0 h_e [B,L,D] 1 h_c [B,L,D] 2 h_share [B,L,D] 3 pos_emb [25,50]
//  4 W_hid [300,650] 5 b_hid [300] 6 ln_g [300] 7 ln_b [300]
//  8 W_rel [1,300] 9 b_rel [1]
// d_out: pair_score [B*NPAIR] floats, then emo_cau_pos [NPAIR*2] as floats.
// ws layout: A (4096*300 f32) | C (4096*300 f32) | pairbase (25*300 f32)
//            | ptab (NPAIR*3 i32)   -> total ~9.44 MiB
// ---------------------------------------------------------------------------
extern "C" void kernel_launch(void* const* d_in, const int* in_sizes, int n_in,
                              void* d_out, int out_size, void* d_ws, size_t ws_size,
                              hipStream_t stream) {
  (void)in_sizes; (void)n_in; (void)out_size; (void)ws_size;

  const float* h_e     = (const float*)d_in[0];
  const float* h_c     = (const float*)d_in[1];
  const float* h_share = (const float*)d_in[2];
  const float* pos_emb = (const float*)d_in[3];
  const float* W_hid   = (const float*)d_in[4];
  const float* b_hid   = (const float*)d_in[5];
  const float* ln_g    = (const float*)d_in[6];
  const float* ln_b    = (const float*)d_in[7];
  const float* W_rel   = (const float*)d_in[8];
  const float* b_rel   = (const float*)d_in[9];

  char* ws = (char*)d_ws;
  float* A_buf    = (float*)(ws);                                    // 4,915,200 B
  float* C_buf    = (float*)(ws + (size_t)MROWS * DD * 4);           // 4,915,200 B
  float* pairbase = (float*)(ws + (size_t)2 * MROWS * DD * 4);       //    30,000 B
  int*   ptab     = (int*)  (ws + (size_t)2 * MROWS * DD * 4 + NREL * DD * 4);

  float* score_out = (float*)d_out;                    // [B*NPAIR]
  float* pos_out   = score_out + (size_t)BB * NPAIR;   // [NPAIR*2]

  // 0) pair table + emo_cau_pos
  build_pairs_kernel<<<1, LL, 0, stream>>>(ptab, pos_out);

  // 1) 25x300 positional base table
  build_pairbase_kernel<<<1, 256, 0, stream>>>(pos_emb, W_hid, b_hid, pairbase);

  // 2) dense GEMMs on the WMMA fp32 pipe: grid (64 mblocks, 1, 2 mats)
  dim3 ggrid(MROWS / 64, 1, 2);
  gemm_wmma_kernel<<<ggrid, 128, 0, stream>>>(h_e, h_c, h_share, W_hid,
                                              A_buf, C_buf);

  // 3) per-pair score: one wave per (b, pair) = 97408 waves, 8 waves/block
  const int total_waves = BB * NPAIR;
  const int blocks = (total_waves * 32 + 255) / 256;
  pair_score_kernel<<<blocks, 256, 0, stream>>>(A_buf, C_buf, pairbase, ptab,
                                                ln_g, ln_b, W_rel, b_rel,
                                                score_out);
}